// EagleNet_85383949844918
// MI455X (gfx1250) — compile-verified
//
#include <hip/hip_runtime.h>
#include <math.h>

// ---------------------------------------------------------------------------
// EagleNet pipeline for MI455X (gfx1250, wave32).
//   K1 embed_gemm      : 9216x1024 @ 1024x256 via V_WMMA_F32_16X16X4_F32 (exact fp32)
//   K2 mase_conv       : fused 1/3/5/7-cube stencils + channel reduce + residual (LDS tiled)
//   K3 ln_att          : LayerNorm (wave-per-row, shfl reductions) + x@w_att (N=2)
//   K4 softmax_stats   : global max, per-row exp-sum, alpha, masked alpha mass
//   K5 alpha_x_partial : v = alpha @ x_ln partial sums (coalesced column-major walk)
//   K6 finalize        : v reduce, logits, top-20 iterative argmax, 3x3 neighbor gather
// ---------------------------------------------------------------------------

typedef __attribute__((ext_vector_type(2))) float v2f;
typedef __attribute__((ext_vector_type(8))) float v8f;

#define DG   96
#define HG   96
#define NPAT (DG * HG)      // 9216
#define KP   1024
#define EMB  256
#define NCLS 2
#define TOPK 20

// ---------------------------- K1: embed GEMM -------------------------------
#define BM 128
#define BN 64
#define BK 32
#define LDA 36   // BK + 4 dwords pad; 144 B row stride (16B aligned)
#define LDB 68   // BN + 4 dwords pad; 272 B row stride (16B aligned)

__global__ __launch_bounds__(256) void embed_gemm(
    const float* __restrict__ A,     // cube  [9216][1024]
    const float* __restrict__ B,     // w_embed [1024][256]
    const float* __restrict__ bias,  // b_embed [256]
    float* __restrict__ C)           // emb [9216][256]
{
    __shared__ float As[BM * LDA];
    __shared__ float Bs[BK * LDB];
    const int tid  = threadIdx.x;
    const int lane = tid & 31;
    const int wave = tid >> 5;            // 0..7 -> 16-row strip each
    const int m0   = blockIdx.y * BM;
    const int n0   = blockIdx.x * BN;
    const int mloc = wave * 16 + (lane & 15);   // A-frag row within tile
    const int koff = (lane >> 4) * 2;           // lanes 0-15 -> K 0,1 ; 16-31 -> K 2,3
    const int ncol = lane & 15;

    v8f acc[4] = {};

    for (int k0 = 0; k0 < KP; k0 += BK) {
        // A tile: 128x32 floats = 1024 float4, 4 per thread
        for (int i = tid; i < (BM * BK / 4); i += 256) {
            int r  = i >> 3;        // / (BK/4)
            int c4 = i & 7;
            float4 v = *(const float4*)(A + (size_t)(m0 + r) * KP + k0 + c4 * 4);
            *(float4*)(&As[r * LDA + c4 * 4]) = v;
        }
        // B tile: 32x64 floats = 512 float4, 2 per thread
        for (int i = tid; i < (BK * BN / 4); i += 256) {
            int r  = i >> 4;        // / (BN/4)
            int c4 = i & 15;
            float4 v = *(const float4*)(B + (size_t)(k0 + r) * EMB + n0 + c4 * 4);
            *(float4*)(&Bs[r * LDB + c4 * 4]) = v;
        }
        __syncthreads();
#pragma unroll
        for (int kk = 0; kk < BK; kk += 4) {
            v2f a = *(const v2f*)(&As[mloc * LDA + kk + koff]);
#pragma unroll
            for (int j = 0; j < 4; ++j) {
                v2f b;
                b.x = Bs[(kk + koff) * LDB + j * 16 + ncol];
                b.y = Bs[(kk + koff + 1) * LDB + j * 16 + ncol];
                acc[j] = __builtin_amdgcn_wmma_f32_16x16x4_f32(
                    false, a, false, b, (short)0, acc[j], false, false);
            }
        }
        __syncthreads();
    }
    // C layout: VGPR i holds row (lane>=16 ? 8 : 0)+i, col = lane&15
#pragma unroll
    for (int j = 0; j < 4; ++j) {
        int n = n0 + j * 16 + ncol;
        float bv = bias[n];
#pragma unroll
        for (int i = 0; i < 8; ++i) {
            int m = m0 + wave * 16 + (lane >> 4) * 8 + i;
            C[(size_t)m * EMB + n] = acc[j][i] + bv;
        }
    }
}

// ---------------------------- K2: MASE conv --------------------------------
#define TD 2
#define TH 2
#define TW 64
#define ILD 72   // padded W stride of the LDS input tile (70 -> 72)

__global__ __launch_bounds__(256) void mase_conv(
    const float* __restrict__ emb,
    const float* __restrict__ w1, const float* __restrict__ b1,
    const float* __restrict__ w3, const float* __restrict__ b3,
    const float* __restrict__ w5, const float* __restrict__ b5,
    const float* __restrict__ w7, const float* __restrict__ b7,
    const float* __restrict__ wr, const float* __restrict__ br,
    float* __restrict__ out)
{
    __shared__ float sIn[8 * 8 * ILD];
    __shared__ float sW7[343 * 8];
    __shared__ float sW5[125 * 8];
    __shared__ float sW3[27 * 8];
    __shared__ float sW1[8];
    __shared__ float sWr[32];
    __shared__ float sB[33];    // b1,b3,b5,b7 (8 each) + convr_b

    const int tid = threadIdx.x;
    // repack weights tap-major: one LDS tap value feeds 8 channel FMAs
    for (int i = tid; i < 343 * 8; i += 256) { int tap = i >> 3, oc = i & 7; sW7[i] = w7[oc * 343 + tap]; }
    for (int i = tid; i < 125 * 8; i += 256) { int tap = i >> 3, oc = i & 7; sW5[i] = w5[oc * 125 + tap]; }
    for (int i = tid; i < 27 * 8;  i += 256) { int tap = i >> 3, oc = i & 7; sW3[i] = w3[oc * 27 + tap]; }
    if (tid < 8)  sW1[tid] = w1[tid];
    if (tid < 32) sWr[tid] = wr[tid];
    if (tid < 8)  { sB[tid] = b1[tid]; sB[8 + tid] = b3[tid]; sB[16 + tid] = b5[tid]; sB[24 + tid] = b7[tid]; }
    if (tid == 0) sB[32] = br[0];

    const int w0 = blockIdx.x * TW;
    const int h0 = blockIdx.y * TH;
    const int d0 = blockIdx.z * TD;

    // halo tile load (zero-padded SAME boundary)
    for (int i = tid; i < 8 * 8 * 70; i += 256) {
        int zw = i % 70;
        int zh = (i / 70) & 7;
        int zd = i / 560;
        int d = d0 + zd - 3, h = h0 + zh - 3, w = w0 + zw - 3;
        float v = 0.f;
        if (d >= 0 && d < DG && h >= 0 && h < HG && w >= 0 && w < EMB)
            v = emb[((size_t)(d * HG + h)) * EMB + w];
        sIn[(zd * 8 + zh) * ILD + zw] = v;
    }
    __syncthreads();

    const int lw = tid & 63;
    const int lh = (tid >> 6) & 1;
    const int ld = tid >> 7;

    float a1[8], a3[8], a5[8], a7[8];
#pragma unroll
    for (int oc = 0; oc < 8; ++oc) {
        a1[oc] = sB[oc]; a3[oc] = sB[8 + oc]; a5[oc] = sB[16 + oc]; a7[oc] = sB[24 + oc];
    }

    for (int kd = -3; kd <= 3; ++kd) {
        for (int kh = -3; kh <= 3; ++kh) {
            const float* row = &sIn[((ld + 3 + kd) * 8 + (lh + 3 + kh)) * ILD + lw + 3];
            const bool dh5 = (kd >= -2 && kd <= 2 && kh >= -2 && kh <= 2);
            const bool dh3 = (kd >= -1 && kd <= 1 && kh >= -1 && kh <= 1);
#pragma unroll
            for (int kw = -3; kw <= 3; ++kw) {
                float v = row[kw];
                int t7 = ((kd + 3) * 7 + (kh + 3)) * 7 + (kw + 3);
                const float* p7 = &sW7[t7 * 8];
#pragma unroll
                for (int oc = 0; oc < 8; ++oc) a7[oc] = fmaf(v, p7[oc], a7[oc]);
                if (dh5 && kw >= -2 && kw <= 2) {
                    int t5 = ((kd + 2) * 5 + (kh + 2)) * 5 + (kw + 2);
                    const float* p5 = &sW5[t5 * 8];
#pragma unroll
                    for (int oc = 0; oc < 8; ++oc) a5[oc] = fmaf(v, p5[oc], a5[oc]);
                }
                if (dh3 && kw >= -1 && kw <= 1) {
                    int t3 = ((kd + 1) * 3 + (kh + 1)) * 3 + (kw + 1);
                    const float* p3 = &sW3[t3 * 8];
#pragma unroll
                    for (int oc = 0; oc < 8; ++oc) a3[oc] = fmaf(v, p3[oc], a3[oc]);
                }
                if (kd == 0 && kh == 0 && kw == 0) {
#pragma unroll
                    for (int oc = 0; oc < 8; ++oc) a1[oc] = fmaf(v, sW1[oc], a1[oc]);
                }
            }
        }
    }

    float y = sB[32];
#pragma unroll
    for (int oc = 0; oc < 8; ++oc) {
        y = fmaf(sWr[oc],      fmaxf(a1[oc], 0.f), y);
        y = fmaf(sWr[8 + oc],  fmaxf(a3[oc], 0.f), y);
        y = fmaf(sWr[16 + oc], fmaxf(a5[oc], 0.f), y);
        y = fmaf(sWr[24 + oc], fmaxf(a7[oc], 0.f), y);
    }
    float center = sIn[((ld + 3) * 8 + (lh + 3)) * ILD + lw + 3];
    y = fmaxf(y, 0.f) + center;

    int d = d0 + ld, h = h0 + lh, w = w0 + lw;
    out[((size_t)(d * HG + h)) * EMB + w] = y;
}

// ------------------------ K3: LayerNorm + head proj ------------------------
__global__ __launch_bounds__(256) void ln_att(
    const float* __restrict__ xin,
    const float* __restrict__ g, const float* __restrict__ b,
    const float* __restrict__ watt, const float* __restrict__ batt,
    float* __restrict__ xout, float* __restrict__ outc)
{
    const int lane = threadIdx.x & 31;
    const int wave = threadIdx.x >> 5;
    const int row  = blockIdx.x * 8 + wave;

    const float* xr = xin + (size_t)row * EMB;
    float4 v0 = *(const float4*)(xr + lane * 8);
    float4 v1 = *(const float4*)(xr + lane * 8 + 4);
    float vals[8] = {v0.x, v0.y, v0.z, v0.w, v1.x, v1.y, v1.z, v1.w};

    float s = 0.f;
#pragma unroll
    for (int i = 0; i < 8; ++i) s += vals[i];
#pragma unroll
    for (int off = 16; off; off >>= 1) s += __shfl_xor(s, off, 32);
    float mu = s * (1.0f / 256.0f);

    float q = 0.f;
#pragma unroll
    for (int i = 0; i < 8; ++i) { float d = vals[i] - mu; q = fmaf(d, d, q); }
#pragma unroll
    for (int off = 16; off; off >>= 1) q += __shfl_xor(q, off, 32);
    float rinv = 1.0f / sqrtf(q * (1.0f / 256.0f) + 1e-6f);

    float4 g0 = *(const float4*)(g + lane * 8);
    float4 g1 = *(const float4*)(g + lane * 8 + 4);
    float4 bb0 = *(const float4*)(b + lane * 8);
    float4 bb1 = *(const float4*)(b + lane * 8 + 4);
    float gv[8] = {g0.x, g0.y, g0.z, g0.w, g1.x, g1.y, g1.z, g1.w};
    float bv[8] = {bb0.x, bb0.y, bb0.z, bb0.w, bb1.x, bb1.y, bb1.z, bb1.w};

    float ys[8];
    float p0 = 0.f, p1 = 0.f;
#pragma unroll
    for (int i = 0; i < 8; ++i) {
        int c = lane * 8 + i;
        float y = fmaf((vals[i] - mu) * rinv, gv[i], bv[i]);
        ys[i] = y;
        p0 = fmaf(y, watt[2 * c], p0);
        p1 = fmaf(y, watt[2 * c + 1], p1);
    }
    float4 o0 = {ys[0], ys[1], ys[2], ys[3]};
    float4 o1 = {ys[4], ys[5], ys[6], ys[7]};
    *(float4*)(xout + (size_t)row * EMB + lane * 8) = o0;
    *(float4*)(xout + (size_t)row * EMB + lane * 8 + 4) = o1;

#pragma unroll
    for (int off = 16; off; off >>= 1) { p0 += __shfl_xor(p0, off, 32); p1 += __shfl_xor(p1, off, 32); }
    if (lane == 0) {
        outc[row * 2]     = p0 + batt[0];
        outc[row * 2 + 1] = p1 + batt[1];
    }
}

// ------------------------- K4: softmax-mass stats --------------------------
__global__ __launch_bounds__(1024) void softmax_stats(
    const float* __restrict__ outc, const int* __restrict__ mask,
    float* __restrict__ sarr, float* __restrict__ alpha, float* __restrict__ scal)
{
    __shared__ float red[1024];
    const int t = threadIdx.x;

    float m = -INFINITY;
    for (int i = t; i < NPAT * NCLS; i += 1024) m = fmaxf(m, outc[i]);
    red[t] = m; __syncthreads();
    for (int off = 512; off; off >>= 1) { if (t < off) red[t] = fmaxf(red[t], red[t + off]); __syncthreads(); }
    float gmax = red[0]; __syncthreads();

    float tot = 0.f;
    for (int r = t; r < NPAT; r += 1024) {
        float sv = expf(outc[2 * r] - gmax) + expf(outc[2 * r + 1] - gmax);
        sarr[r] = sv;
        tot += sv;
    }
    red[t] = tot; __syncthreads();
    for (int off = 512; off; off >>= 1) { if (t < off) red[t] += red[t + off]; __syncthreads(); }
    float S = red[0]; __syncthreads();
    float inv = 1.0f / S;

    float am = 0.f;
    for (int r = t; r < NPAT; r += 1024) {
        float al = sarr[r] * inv;
        alpha[r] = al;
        if (mask[r] == 0) am += al;
    }
    red[t] = am; __syncthreads();
    for (int off = 512; off; off >>= 1) { if (t < off) red[t] += red[t + off]; __syncthreads(); }
    if (t == 0) { scal[0] = gmax; scal[1] = S; scal[2] = red[0]; }
}

// --------------------------- K5: v = alpha @ x -----------------------------
__global__ __launch_bounds__(256) void alpha_x_partial(
    const float* __restrict__ alpha, const float* __restrict__ x,
    float* __restrict__ part)   // [144][256]
{
    const int c  = threadIdx.x;
    const int r0 = blockIdx.x * 64;
    float acc = 0.f;
    for (int i = 0; i < 64; ++i)
        acc = fmaf(alpha[r0 + i], x[(size_t)(r0 + i) * EMB + c], acc);
    part[blockIdx.x * EMB + c] = acc;
}

// ------------------ K6: logits, top-k, neighbor gather ---------------------
__global__ __launch_bounds__(1024) void finalize_k(
    const float* __restrict__ part, const float* __restrict__ watt,
    const float* __restrict__ batt, const float* __restrict__ alpha,
    const float* __restrict__ outc, const float* __restrict__ scal,
    float* __restrict__ out)
{
    __shared__ float sv[EMB];
    __shared__ float sAlpha[NPAT];
    __shared__ float rv[1024];
    __shared__ int   ri[1024];
    __shared__ int   topk[TOPK];
    const int t = threadIdx.x;

    if (t < EMB) {
        float acc = 0.f;
        for (int i = 0; i < 144; ++i) acc += part[i * EMB + t];
        sv[t] = acc;
    }
    for (int i = t; i < NPAT; i += 1024) sAlpha[i] = alpha[i];
    __syncthreads();

    if (t == 0) {
        float l0 = batt[0], l1 = batt[1];
        for (int c = 0; c < EMB; ++c) {
            l0 = fmaf(sv[c], watt[2 * c], l0);
            l1 = fmaf(sv[c], watt[2 * c + 1], l1);
        }
        out[0] = l0; out[1] = l1; out[2] = scal[2];
    }

    for (int kIt = 0; kIt < TOPK; ++kIt) {
        float bvv = -INFINITY; int bi = 0x7fffffff;
        for (int i = t; i < NPAT; i += 1024) {
            float v = sAlpha[i];
            if (v > bvv || (v == bvv && i < bi)) { bvv = v; bi = i; }
        }
        rv[t] = bvv; ri[t] = bi; __syncthreads();
        for (int off = 512; off; off >>= 1) {
            if (t < off) {
                float ov = rv[t + off]; int oi = ri[t + off];
                if (ov > rv[t] || (ov == rv[t] && oi < ri[t])) { rv[t] = ov; ri[t] = oi; }
            }
            __syncthreads();
        }
        if (t == 0) { topk[kIt] = ri[0]; sAlpha[ri[0]] = -INFINITY; }
        __syncthreads();
    }

    if (t < TOPK * 9) {
        const int relr[9] = {-1, -1, -1, 0, 0, 0, 1, 1, 1};
        const int relc[9] = {-1, 0, 1, -1, 0, 1, -1, 0, 1};
        int ti = t / 9, j = t % 9;
        int idx = topk[ti];
        int row = idx / HG, col = idx % HG;
        int nr = row + relr[j], nc = col + relc[j];
        bool valid = (nr >= 0 && nr < DG && nc >= 0 && nc < HG);
        int cr = nr < 0 ? 0 : (nr > DG - 1 ? DG - 1 : nr);
        int cc = nc < 0 ? 0 : (nc > HG - 1 ? HG - 1 : nc);
        int lin = cr * HG + cc;
        float f = valid ? 1.f : 0.f;
        out[3 + 2 * t]     = outc[2 * lin] * f;
        out[3 + 2 * t + 1] = outc[2 * lin + 1] * f;
        out[3 + 2 * TOPK * 9 + t] = alpha[lin] * f;
    }
}

// ------------------------------- launcher ----------------------------------
extern "C" void kernel_launch(void* const* d_in, const int* in_sizes, int n_in,
                              void* d_out, int out_size, void* d_ws, size_t ws_size,
                              hipStream_t stream) {
    const float* cube    = (const float*)d_in[0];
    const int*   mask    = (const int*)  d_in[1];
    const float* w_embed = (const float*)d_in[2];
    const float* b_embed = (const float*)d_in[3];
    const float* conv1_w = (const float*)d_in[4];
    const float* conv1_b = (const float*)d_in[5];
    const float* conv3_w = (const float*)d_in[6];
    const float* conv3_b = (const float*)d_in[7];
    const float* conv5_w = (const float*)d_in[8];
    const float* conv5_b = (const float*)d_in[9];
    const float* conv7_w = (const float*)d_in[10];
    const float* conv7_b = (const float*)d_in[11];
    const float* convr_w = (const float*)d_in[12];
    const float* convr_b = (const float*)d_in[13];
    const float* ln_g    = (const float*)d_in[14];
    const float* ln_b    = (const float*)d_in[15];
    const float* w_att   = (const float*)d_in[16];
    const float* b_att   = (const float*)d_in[17];

    float* ws = (float*)d_ws;
    const size_t embOff   = 0;                         // emb, later reused as x_ln
    const size_t convOff  = embOff  + (size_t)NPAT * EMB;
    const size_t outcOff  = convOff + (size_t)NPAT * EMB;
    const size_t sOff     = outcOff + (size_t)NPAT * NCLS;
    const size_t alphaOff = sOff    + NPAT;
    const size_t partOff  = alphaOff + NPAT;
    const size_t scalOff  = partOff + (size_t)144 * EMB;

    embed_gemm<<<dim3(EMB / BN, NPAT / BM), 256, 0, stream>>>(
        cube, w_embed, b_embed, ws + embOff);

    mase_conv<<<dim3(EMB / TW, HG / TH, DG / TD), 256, 0, stream>>>(
        ws + embOff, conv1_w, conv1_b, conv3_w, conv3_b, conv5_w, conv5_b,
        conv7_w, conv7_b, convr_w, convr_b, ws + convOff);

    ln_att<<<dim3(NPAT / 8), 256, 0, stream>>>(
        ws + convOff, ln_g, ln_b, w_att, b_att, ws + embOff, ws + outcOff);

    softmax_stats<<<1, 1024, 0, stream>>>(
        ws + outcOff, mask, ws + sOff, ws + alphaOff, ws + scalOff);

    alpha_x_partial<<<dim3(NPAT / 64), 256, 0, stream>>>(
        ws + alphaOff, ws + embOff, ws + partOff);

    finalize_k<<<1, 1024, 0, stream>>>(
        ws + partOff, w_att, b_att, ws + alphaOff, ws + outcOff, ws + scalOff,
        (float*)d_out);
}